// SmoothRegLoss_86672440033387
// MI455X (gfx1250) — compile-verified
//
#include <hip/hip_runtime.h>
#include <math.h>

#define BATCH 128

typedef __attribute__((ext_vector_type(2))) float v2f;
typedef __attribute__((ext_vector_type(8))) float v8f;

#if __has_builtin(__builtin_amdgcn_wmma_f32_16x16x4_f32)
#define HAVE_WMMA_F32X4 1
#else
#define HAVE_WMMA_F32X4 0
#endif

// ---------------------------------------------------------------- init
__global__ __launch_bounds__(256)
void init_kernel(float* __restrict__ deg, unsigned* __restrict__ maxc,
                 unsigned* __restrict__ maxnc, float* __restrict__ iso, int V) {
  int i = blockIdx.x * blockDim.x + threadIdx.x;
  if (i < V) deg[i] = 0.0f;
  if (i < BATCH) { maxc[i] = 0u; maxnc[i] = 0u; iso[i] = 0.0f; }
}

// ------------------------------------------- sigmoid + transpose to [V,B]
// pred is [B,V]; pcT/propT are [V,B] so the 128 batch values of one vertex
// are contiguous (512B) -> coalesced gathers/atomics in the scatter kernel.
__global__ __launch_bounds__(256)
void sig_transpose_kernel(const float* __restrict__ pred, float* __restrict__ pcT,
                          float* __restrict__ propT, int V) {
  __shared__ float tile[32][33];
  int v0 = blockIdx.x * 32;
  int b0 = blockIdx.y * 32;
  #pragma unroll
  for (int j = 0; j < 4; ++j) {
    int b = b0 + threadIdx.y + j * 8;
    int v = v0 + threadIdx.x;
    float x = (v < V) ? pred[(size_t)b * V + v] : 0.0f;
    tile[threadIdx.y + j * 8][threadIdx.x] = 1.0f / (1.0f + expf(-x));
  }
  __syncthreads();
  #pragma unroll
  for (int j = 0; j < 4; ++j) {
    int v = v0 + threadIdx.y + j * 8;
    int b = b0 + threadIdx.x;
    if (v < V) {
      float s = tile[threadIdx.x][threadIdx.y + j * 8];
      pcT[(size_t)v * BATCH + b]   = s;   // sigmoid(pred)^T
      propT[(size_t)v * BATCH + b] = s;   // propagate identity term
    }
  }
}

// ---------------------------------------------------------------- degree
// row = [f0,f1,f2,f1,f2,f0]: each face contributes 2 to each of its vertices.
__global__ __launch_bounds__(256)
void degree_kernel(const int* __restrict__ faces, float* __restrict__ deg, int F) {
  int f = blockIdx.x * blockDim.x + threadIdx.x;
  if (f >= F) return;
  atomicAdd(&deg[faces[3 * f + 0]], 2.0f);
  atomicAdd(&deg[faces[3 * f + 1]], 2.0f);
  atomicAdd(&deg[faces[3 * f + 2]], 2.0f);
}

// ---------------------------------------------------------------- scatter
// Six (row,col) entries per face collapse to 3 atomics of pair-sums.
// Lane b handles batch b; 128 consecutive threads share one face.
__global__ __launch_bounds__(256)
void scatter_kernel(const int* __restrict__ faces, const float* __restrict__ pcT,
                    float* __restrict__ propT, int F) {
  int tid = blockIdx.x * blockDim.x + threadIdx.x;
  int b = tid & (BATCH - 1);
  int f = tid >> 7;
  if (f >= F) return;
  int f0 = faces[3 * f + 0];
  int f1 = faces[3 * f + 1];
  int f2 = faces[3 * f + 2];
  float x0 = pcT[(size_t)f0 * BATCH + b];
  float x1 = pcT[(size_t)f1 * BATCH + b];
  float x2 = pcT[(size_t)f2 * BATCH + b];
  atomicAdd(&propT[(size_t)f0 * BATCH + b], x1 + x2);
  atomicAdd(&propT[(size_t)f1 * BATCH + b], x0 + x2);
  atomicAdd(&propT[(size_t)f2 * BATCH + b], x0 + x1);
}

// ------------------------------------------------------------- per-batch max
// prop_c and prop_nc are >= 0, so float bits are order-preserving as uint.
__global__ __launch_bounds__(256)
void max_kernel(const float* __restrict__ propT, const float* __restrict__ deg,
                unsigned* __restrict__ maxcB, unsigned* __restrict__ maxncB, int V) {
  int stride = gridDim.x * blockDim.x;       // multiple of 128 -> b fixed per thread
  int idx = blockIdx.x * blockDim.x + threadIdx.x;
  int b = idx & (BATCH - 1);
  float mc = 0.0f, mnc = 0.0f;
  int total = V * BATCH;
  for (int i = idx; i < total; i += stride) {
    int v = i >> 7;
    float pc_ = propT[i];
    float pnc = (1.0f + deg[v]) - pc_;       // prop_nc via linearity
    mc  = fmaxf(mc,  pc_);
    mnc = fmaxf(mnc, pnc);
  }
  atomicMax(&maxcB[b],  __float_as_uint(mc));
  atomicMax(&maxncB[b], __float_as_uint(mnc));
}

// ------------------------------------------------------------- iso reduction
// Unchecked per-element term; reciprocals hoisted (invc/invnc) so the loop
// body is pure mul/sub/abs -> no div sequence, no divergence.
__device__ __forceinline__
float iso_term_u(const float* __restrict__ pcT, const float* __restrict__ propT,
                 const float* __restrict__ deg, int v, int b,
                 float invc, float invnc) {
  size_t idx = (size_t)v * BATCH + b;
  float pc     = pcT[idx];
  float propc  = propT[idx];
  float propnc = (1.0f + deg[v]) - propc;
  return fabsf(pc - propc * invc) + fabsf((1.0f - pc) - propnc * invnc);
}

// Each wave owns a 16-batch column group (8 waves x 16 = 128 batches/block).
// WMMA: D = ones(16x4) x B(4x16) + C  => every row of D holds column sums of B.
// B operand: lane%16 = N (batch), VGPR index + lane-half = K (vertex row);
// all 4 K rows sum into the same column, so exact K slotting is irrelevant.
// D extraction uses the documented C/D layout: VGPR0, lanes 0-15: M=0, N=lane.
__global__ __launch_bounds__(256)
void iso_kernel(const float* __restrict__ pcT, const float* __restrict__ propT,
                const float* __restrict__ deg, const unsigned* __restrict__ maxcB,
                const unsigned* __restrict__ maxncB, float* __restrict__ iso, int V) {
  const int lane  = threadIdx.x & 31;
  const int wave  = threadIdx.x >> 5;          // 0..7
  const int bbase = wave * 16;
  const int b     = bbase + (lane & 15);
  const float invc  = 1.0f / (__uint_as_float(maxcB[b])  + 1e-6f);
  const float invnc = 1.0f / (__uint_as_float(maxncB[b]) + 1e-6f);
  const int koff = (lane < 16) ? 0 : 2;        // which K rows this lane half feeds
  const int step = gridDim.x * 4;

#if HAVE_WMMA_F32X4
  v8f acc0 = {0.f, 0.f, 0.f, 0.f, 0.f, 0.f, 0.f, 0.f};
  v8f acc1 = acc0;
  v2f ones; ones.x = 1.0f; ones.y = 1.0f;

  int v0 = blockIdx.x * 4;
  // main loop: two independent WMMA accumulation chains, no bounds checks
  for (; v0 + step + 3 < V; v0 += 2 * step) {
    int ra = v0 + koff;
    int rb = v0 + step + koff;
    v2f bma, bmb;
    bma.x = iso_term_u(pcT, propT, deg, ra,     b, invc, invnc);
    bma.y = iso_term_u(pcT, propT, deg, ra + 1, b, invc, invnc);
    bmb.x = iso_term_u(pcT, propT, deg, rb,     b, invc, invnc);
    bmb.y = iso_term_u(pcT, propT, deg, rb + 1, b, invc, invnc);
    acc0 = __builtin_amdgcn_wmma_f32_16x16x4_f32(false, ones, false, bma,
                                                 (short)0, acc0, false, false);
    acc1 = __builtin_amdgcn_wmma_f32_16x16x4_f32(false, ones, false, bmb,
                                                 (short)0, acc1, false, false);
  }
  // tail: masked, branch-free per lane (EXEC stays full for WMMA)
  for (; v0 < V; v0 += step) {
    int r0 = v0 + koff;
    int r1 = r0 + 1;
    int c0 = (r0 < V) ? r0 : (V - 1);
    int c1 = (r1 < V) ? r1 : (V - 1);
    float m0 = (r0 < V) ? 1.0f : 0.0f;
    float m1 = (r1 < V) ? 1.0f : 0.0f;
    v2f bm;
    bm.x = m0 * iso_term_u(pcT, propT, deg, c0, b, invc, invnc);
    bm.y = m1 * iso_term_u(pcT, propT, deg, c1, b, invc, invnc);
    acc0 = __builtin_amdgcn_wmma_f32_16x16x4_f32(false, ones, false, bm,
                                                 (short)0, acc0, false, false);
  }
  if (lane < 16) atomicAdd(&iso[b], acc0[0] + acc1[0]);  // D row M=0 = col sums
#else
  float acc = 0.0f;
  for (int v0 = blockIdx.x * 4; v0 < V; v0 += step) {
    int r0 = v0 + koff;
    int r1 = r0 + 1;
    if (r0 < V) acc += iso_term_u(pcT, propT, deg, r0, b, invc, invnc);
    if (r1 < V) acc += iso_term_u(pcT, propT, deg, r1, b, invc, invnc);
  }
  atomicAdd(&iso[b], acc);
#endif
}

// ---------------------------------------------------------------- final mean
__global__ __launch_bounds__(BATCH)
void final_kernel(const float* __restrict__ iso, float* __restrict__ out, float norm) {
  __shared__ float red[BATCH];
  int t = threadIdx.x;
  red[t] = log1pf(iso[t] / norm);
  __syncthreads();
  for (int s = BATCH / 2; s > 0; s >>= 1) {
    if (t < s) red[t] += red[t + s];
    __syncthreads();
  }
  if (t == 0) out[0] = red[0] / (float)BATCH;
}

extern "C" void kernel_launch(void* const* d_in, const int* in_sizes, int n_in,
                              void* d_out, int out_size, void* d_ws, size_t ws_size,
                              hipStream_t stream) {
  const float* pred  = (const float*)d_in[0];
  const int*   faces = (const int*)d_in[1];
  float*       out   = (float*)d_out;
  const int V = in_sizes[0] / BATCH;
  const int F = in_sizes[1] / 3;

  // Workspace layout (~51.5 MB total)
  float*    pcT   = (float*)d_ws;                       // V*128
  float*    propT = pcT + (size_t)V * BATCH;            // V*128
  float*    deg   = propT + (size_t)V * BATCH;          // V
  unsigned* maxc  = (unsigned*)(deg + V);               // 128
  unsigned* maxnc = maxc + BATCH;                       // 128
  float*    iso   = (float*)(maxnc + BATCH);            // 128

  init_kernel<<<(V + 255) / 256, 256, 0, stream>>>(deg, maxc, maxnc, iso, V);

  dim3 tb(32, 8);
  dim3 tg((V + 31) / 32, BATCH / 32);
  sig_transpose_kernel<<<tg, tb, 0, stream>>>(pred, pcT, propT, V);

  degree_kernel<<<(F + 255) / 256, 256, 0, stream>>>(faces, deg, F);

  long long sthreads = (long long)F * BATCH;
  scatter_kernel<<<(unsigned)((sthreads + 255) / 256), 256, 0, stream>>>(faces, pcT, propT, F);

  max_kernel<<<1024, 256, 0, stream>>>(propT, deg, maxc, maxnc, V);

  int isoBlocks = (V + 3) / 4; if (isoBlocks > 1024) isoBlocks = 1024;
  iso_kernel<<<isoBlocks, 256, 0, stream>>>(pcT, propT, deg, maxc, maxnc, iso, V);

  // norm_factor = sum(prop_c)+sum(prop_nc) = sum_v(1+deg[v]) = V + 6F (exact)
  float norm = (float)((double)V + 6.0 * (double)F) + 1e-3f;
  final_kernel<<<1, BATCH, 0, stream>>>(iso, out, norm);
}